// LandmarkLoss_73778948210832
// MI455X (gfx1250) — compile-verified
//
#include <hip/hip_runtime.h>

// Problem constants from the reference: B=16, N=4096, W=1024, H=1024
#define BB 16
#define NN 4096
#define WW 1024
#define HH 1024
#define WH (WW * HH)

typedef float v2f __attribute__((ext_vector_type(2)));
typedef float v8f __attribute__((ext_vector_type(8)));

// ---------------------------------------------------------------------------
// Pass 1: one thread per landmark point. Coalesced float4 landmark load,
// 8 scattered b32 gathers (y-neighbors share cachelines), faithful weight
// formula, then wave32 shuffle + LDS reduction to one partial per block.
// ---------------------------------------------------------------------------
__global__ __launch_bounds__(256) void landmark_partial_kernel(
    const float* __restrict__ lm,    // [B,N,4]
    const float* __restrict__ flow,  // [B,2,W,H]
    float* __restrict__ partial)     // [gridDim.x]
{
    const int tid = blockIdx.x * blockDim.x + threadIdx.x;  // 0 .. B*N-1
    const int b = tid >> 12;                                // tid / N (N=4096)

    const float4 l = ((const float4*)lm)[tid];
    const float x1 = l.x, y1 = l.y, x2 = l.z, y2 = l.w;

    const float x1d = floorf(x1), y1d = floorf(y1);
    const float x1u = x1d + 1.0f, y1u = y1d + 1.0f;
    const bool valid = (x1u < (float)WW) && (y1u < (float)HH);

    int xd = (int)x1d; xd = xd < 0 ? 0 : (xd > WW - 2 ? WW - 2 : xd);
    int yd = (int)y1d; yd = yd < 0 ? 0 : (yd > HH - 2 ? HH - 2 : yd);

    // flow[b, c, x, y] at ((b*2 + c)*W + x)*H + y
    const float* f0 = flow + (long)(b * 2) * WH + (long)xd * HH + yd;
    const float* f1 = f0 + WH;

    // corners: a=(xd,yd)  d=(xd,yu)  c=(xu,yd)  b=(xu,yu)
    const float a0 = f0[0], d0 = f0[1], c0 = f0[HH], b0 = f0[HH + 1];
    const float a1 = f1[0], d1 = f1[1], c1 = f1[HH], b1 = f1[HH + 1];

    const float fx = x1 - x1d;
    const float fy = y1 - y1d;
    const float gx = x1u - x1;
    const float gy = y1u - x1;   // faithful to reference: uses x1, not y1

    const float wa = fx * fy;
    const float wb = gx * gy;
    const float wc = gx * fy;
    const float wd = fx * gy;

    const float o0 = a0 * wa + b0 * wb + c0 * wc + d0 * wd;
    const float o1 = a1 * wa + b1 * wb + c1 * wc + d1 * wd;

    const float ex = x1 + o0 - x2;
    const float ey = y1 + o1 - y2;
    float per = valid ? (ex * ex + ey * ey) : 0.0f;

    // wave32 reduction
    #pragma unroll
    for (int off = 16; off > 0; off >>= 1)
        per += __shfl_xor(per, off, 32);

    __shared__ float sm[8];
    const int lane = threadIdx.x & 31;
    const int wave = threadIdx.x >> 5;
    if (lane == 0) sm[wave] = per;
    __syncthreads();

    if (wave == 0) {
        float s = (lane < 8) ? sm[lane] : 0.0f;
        s += __shfl_xor(s, 4, 32);
        s += __shfl_xor(s, 2, 32);
        s += __shfl_xor(s, 1, 32);
        if (lane == 0) partial[blockIdx.x] = s;
    }
}

// ---------------------------------------------------------------------------
// Pass 2: single wave32 reduces 256 partials through the matrix pipe.
// A = all-ones 16x4 f32  =>  D[m,n] = sum_k B[k,n]  (layout-independent total).
// Four chained V_WMMA_F32_16X16X4_F32 accumulate all 256 values into column
// sums; every row of D holds the same 16 column sums (lanes 0-15 and 16-31
// each carry one full copy in VGPR0), so a full 32-lane shuffle-sum of acc[0]
// yields 2 * total. Fixed operation order -> deterministic across replays.
// loss = total / (2*B) = (2*total) / 64.
// ---------------------------------------------------------------------------
__global__ __launch_bounds__(32) void wmma_reduce_kernel(
    const float* __restrict__ partial,  // [256]
    float* __restrict__ out)            // [1]
{
    const int lane = threadIdx.x;  // 0..31, EXEC all ones (required for WMMA)

    v2f ones;
    ones.x = 1.0f;
    ones.y = 1.0f;

    v8f acc = {};
    #pragma unroll
    for (int j = 0; j < 4; ++j) {
        v2f bvec;
        bvec.x = partial[j * 64 + lane * 2 + 0];
        bvec.y = partial[j * 64 + lane * 2 + 1];
        // 8 args: (neg_a, A, neg_b, B, c_mod, C, reuse_a, reuse_b)
        acc = __builtin_amdgcn_wmma_f32_16x16x4_f32(
            /*neg_a=*/false, ones, /*neg_b=*/false, bvec,
            /*c_mod=*/(short)0, acc, /*reuse_a=*/false, /*reuse_b=*/false);
    }

    float t = acc[0];
    #pragma unroll
    for (int off = 16; off > 0; off >>= 1)
        t += __shfl_xor(t, off, 32);

    if (lane == 0)
        out[0] = t * (1.0f / 64.0f);  // (2*total) * 1/64 = total / 32 = total/(2B)
}

extern "C" void kernel_launch(void* const* d_in, const int* in_sizes, int n_in,
                              void* d_out, int out_size, void* d_ws, size_t ws_size,
                              hipStream_t stream) {
    const float* lm   = (const float*)d_in[0];  // landmarks [B,N,4] f32
    const float* flow = (const float*)d_in[1];  // flow [B,2,W,H] f32
    float* out = (float*)d_out;                 // scalar loss
    float* ws  = (float*)d_ws;                  // 256 partial sums

    const int total = BB * NN;        // 65536 points
    const int block = 256;
    const int grid  = total / block;  // 256 blocks

    landmark_partial_kernel<<<grid, block, 0, stream>>>(lm, flow, ws);
    wmma_reduce_kernel<<<1, 32, 0, stream>>>(ws, out);
}